// Sent_DAN_Simple_75849122448043
// MI455X (gfx1250) — compile-verified
//
#include <hip/hip_runtime.h>
#include <hip/hip_bf16.h>

typedef __attribute__((ext_vector_type(2))) float v2f;
typedef __attribute__((ext_vector_type(8))) float v8f;

#define B_   128
#define S_   2048
#define D_   768
#define NCHUNK 16
#define SCHUNK (S_ / NCHUNK)   // 128

// ---------------------------------------------------------------------------
// Kernel 1: mask floats + per-batch token counts (counts = max(sum,1))
// ---------------------------------------------------------------------------
__global__ __launch_bounds__(256) void mask_count_kernel(
    const int* __restrict__ ids, float* __restrict__ maskf, float* __restrict__ counts) {
  const int b = blockIdx.x;
  const int t = threadIdx.x;
  float cnt = 0.0f;
  for (int s = t; s < S_; s += 256) {
    float m = (ids[(size_t)b * S_ + s] != 0) ? 1.0f : 0.0f;
    maskf[(size_t)b * S_ + s] = m;
    cnt += m;
  }
  __shared__ float red[256];
  red[t] = cnt;
  __syncthreads();
  for (int off = 128; off > 0; off >>= 1) {
    if (t < off) red[t] += red[t + off];
    __syncthreads();
  }
  if (t == 0) counts[b] = fmaxf(red[0], 1.0f);
}

// ---------------------------------------------------------------------------
// Kernel 2: bandwidth phase. Each block = (s-chunk c, batch b); 192 threads
// cover the full D=768 row as float4. Streams 805MB of `inputs` exactly once.
// Deterministic partial sums (no float atomics).
// ---------------------------------------------------------------------------
__global__ __launch_bounds__(192) void pool_partial_kernel(
    const float* __restrict__ in, const float* __restrict__ maskf,
    float* __restrict__ partial) {
  const int c = blockIdx.x;   // 0..15  s-chunk
  const int b = blockIdx.y;   // 0..127 batch
  const int t = threadIdx.x;  // 0..191 -> float4 lane over D
  const int s0 = c * SCHUNK;

  const float* base = in + ((size_t)b * S_ + s0) * D_;
  const float* mb   = maskf + (size_t)b * S_ + s0;

  float4 acc = make_float4(0.f, 0.f, 0.f, 0.f);
  for (int s = 0; s < SCHUNK; ++s) {
    const float  m = mb[s];
    const float4 v = ((const float4*)(base + (size_t)s * D_))[t];
    acc.x += m * v.x;
    acc.y += m * v.y;
    acc.z += m * v.z;
    acc.w += m * v.w;
  }
  ((float4*)(partial + ((size_t)c * B_ + b) * D_))[t] = acc;
}

// ---------------------------------------------------------------------------
// Kernel 3: reduce the 16 partials, divide by count -> pooled[b, d]
// ---------------------------------------------------------------------------
__global__ __launch_bounds__(192) void pool_reduce_kernel(
    const float* __restrict__ partial, const float* __restrict__ counts,
    float* __restrict__ pooled) {
  const int b = blockIdx.x;
  const int t = threadIdx.x;
  float4 acc = make_float4(0.f, 0.f, 0.f, 0.f);
  for (int c = 0; c < NCHUNK; ++c) {
    const float4 v = ((const float4*)(partial + ((size_t)c * B_ + b) * D_))[t];
    acc.x += v.x; acc.y += v.y; acc.z += v.z; acc.w += v.w;
  }
  const float inv = 1.0f / counts[b];
  float4 r = make_float4(acc.x * inv, acc.y * inv, acc.z * inv, acc.w * inv);
  ((float4*)(pooled + (size_t)b * D_))[t] = r;
}

// ---------------------------------------------------------------------------
// Kernel 4: h = pooled @ W1^T + b1 via V_WMMA_F32_16X16X4_F32.
// One wave32 per 16x16 output tile; K=768 in steps of 4.
// A (16x4 f32): lane half -> K pair (0,1)/(2,3). B (4x16): B[kk][n]=W1[n][k+kk]
// has the symmetric layout, so both sides load a float2 at k + 2*(lane>>4).
// D layout: VGPR r -> row m0 + r + 8*(lane>>4), col n0 + (lane&15).
// ---------------------------------------------------------------------------
__global__ __launch_bounds__(32) void gemm1_wmma_kernel(
    const float* __restrict__ pooled, const float* __restrict__ W1,
    const float* __restrict__ b1, float* __restrict__ h) {
  const int n0   = blockIdx.x * 16;
  const int m0   = blockIdx.y * 16;
  const int lane = threadIdx.x;      // 0..31, EXEC all ones (WMMA requirement)
  const int half = lane >> 4;        // 0 or 1
  const int lr   = lane & 15;

  v8f acc = {};
  for (int k = 0; k < D_; k += 4) {
    const int ka = k + 2 * half;
    v2f a = *(const v2f*)(pooled + (size_t)(m0 + lr) * D_ + ka);
    v2f b = *(const v2f*)(W1     + (size_t)(n0 + lr) * D_ + ka);
    acc = __builtin_amdgcn_wmma_f32_16x16x4_f32(
        /*neg_a=*/false, a, /*neg_b=*/false, b,
        /*c_mod=*/(short)0, acc, /*reuse_a=*/false, /*reuse_b=*/false);
  }

  const float bias = b1[n0 + lr];
#pragma unroll
  for (int r = 0; r < 8; ++r) {
    h[(size_t)(m0 + r + 8 * half) * D_ + n0 + lr] = acc[r] + bias;
  }
}

// ---------------------------------------------------------------------------
// Kernel 5: out[b, c] = h[b,:] . W2[c,:] + b2[c]   (256 dot products of 768)
// ---------------------------------------------------------------------------
__global__ __launch_bounds__(256) void head_kernel(
    const float* __restrict__ h, const float* __restrict__ W2,
    const float* __restrict__ b2, float* __restrict__ out) {
  const int t = threadIdx.x;          // 0..255
  const int b = t >> 1;
  const int c = t & 1;
  float acc = b2[c];
  const float* hr = h  + (size_t)b * D_;
  const float* wr = W2 + (size_t)c * D_;
  for (int k = 0; k < D_; ++k) acc += hr[k] * wr[k];
  out[(size_t)b * 2 + c] = acc;
}

// ---------------------------------------------------------------------------
extern "C" void kernel_launch(void* const* d_in, const int* in_sizes, int n_in,
                              void* d_out, int out_size, void* d_ws, size_t ws_size,
                              hipStream_t stream) {
  const float* inputs = (const float*)d_in[0];
  const int*   ids    = (const int*)d_in[1];
  const float* W1     = (const float*)d_in[2];
  const float* b1     = (const float*)d_in[3];
  const float* W2     = (const float*)d_in[4];
  const float* b2     = (const float*)d_in[5];
  float*       out    = (float*)d_out;

  // Workspace layout (all 16B aligned):
  char*  ws      = (char*)d_ws;
  float* maskf   = (float*)ws;                                     // 128*2048 f32 = 1 MB
  float* counts  = (float*)(ws + (size_t)B_ * S_ * 4);             // 128 f32 (+pad)
  float* partial = (float*)(ws + (size_t)B_ * S_ * 4 + 1024);      // 16*128*768 f32 = 6 MB
  float* pooled  = partial + (size_t)NCHUNK * B_ * D_;             // 128*768 f32
  float* h       = pooled  + (size_t)B_ * D_;                      // 128*768 f32

  mask_count_kernel <<<B_, 256, 0, stream>>>(ids, maskf, counts);
  pool_partial_kernel<<<dim3(NCHUNK, B_), 192, 0, stream>>>(inputs, maskf, partial);
  pool_reduce_kernel<<<B_, 192, 0, stream>>>(partial, counts, pooled);
  gemm1_wmma_kernel  <<<dim3(D_ / 16, B_ / 16), 32, 0, stream>>>(pooled, W1, b1, h);
  head_kernel        <<<1, 256, 0, stream>>>(h, W2, b2, out);
}